// FusedColonyWorldModelBridge_43224550867172
// MI455X (gfx1250) — compile-verified
//
#include <hip/hip_runtime.h>

// ---------------------------------------------------------------------------
// FusedColonyWorldModelBridge: E8 two-level residual quantizer + Fano octonion
// alignment, for MI455X (gfx1250, wave32, WMMA).
//
// Outputs concatenated flat (all written as float):
//   quantized_ste : [0,           3670016)   (B*7*8)
//   indices       : [3670016,     4587520)   (2*B*7), written as float(idx)
//   q_err         : [4587520,     4587521)
//   products      : [4587521,     8257537)   (B*7*8)
//   alignments    : [8257537,     8716289)   (B*7)
//   fano_loss     : [8716289,     8716290)
// ---------------------------------------------------------------------------

typedef float v2f __attribute__((ext_vector_type(2)));
typedef float v8f __attribute__((ext_vector_type(8)));

#define NROOTS    240
#define NROWS     458752              // 65536 * 7
#define QUANT_OFF 0
#define IDX_OFF   3670016
#define QERR_OFF  4587520
#define PROD_OFF  4587521
#define ALIGN_OFF 8257537
#define FANO_OFF  8716289

#define K1_BLOCKS 512                 // 512 blocks * 8 waves = 4096 waves
#define K1_WAVES  4096
#define K1_ITERS  7                   // 28672 row-tiles / 4096 waves
#define K2_BLOCKS 1792                // 458752 / 256

#define TILE_CHUNK 3                  // concurrent accumulator chains per chunk
#define NCHUNKS    5                  // 5 * 3 = 15 codebook column-tiles

// Order-preserving float -> uint mapping (f1 < f2  <=>  key(f1) < key(f2)).
__device__ __forceinline__ unsigned int fkey(float f)
{
    unsigned int u = __float_as_uint(f);
    return (u & 0x80000000u) ? ~u : (u | 0x80000000u);
}

// ---------------------------------------------------------------------------
// Kernel 1: two-level E8 quantization via V_WMMA_F32_16X16X4_F32.
// One wave handles a 16-row tile of x_flat; K=8 done as two chained K=4 WMMAs.
// Codebook (B-fragments + r2) stays in LDS; each chunk of 3 tiles loads its
// fragments (1 ds_load_2addr_b64 each), runs 6 WMMAs over 3 independent
// accumulator chains, and folds into the running argmin. sched_barrier fences
// keep accumulator lifetimes short so allocation stays under 256 VGPRs.
// ---------------------------------------------------------------------------
__global__ __launch_bounds__(256)
void quant_kernel(const float* __restrict__ x, const float* __restrict__ roots,
                  float* __restrict__ out, float* __restrict__ ws)
{
    __shared__ float rootsS[NROOTS * 8];
    __shared__ float r2S[NROOTS];
    __shared__ float xS[8][128];   // per-wave 16x8 x tile
    __shared__ float qS[8][128];   // per-wave qsum
    __shared__ float rS[8][128];   // per-wave residual
    __shared__ int   idxS[8][16];
    __shared__ float redS[256];

    const int tid = threadIdx.x;

    // Stage codebook + squared norms into LDS (shared by all 8 waves).
    for (int n = tid; n < NROOTS; n += 256) {
        float s = 0.f;
        #pragma unroll
        for (int k = 0; k < 8; ++k) {
            float v = roots[n * 8 + k];
            rootsS[n * 8 + k] = v;
            s += v * v;
        }
        r2S[n] = s;
    }
    __syncthreads();

    const int lane = tid & 31;
    const int wv   = tid >> 5;
    const int col  = lane & 15;      // M for A-frag, N for B-frag/accumulator
    const int half = lane >> 4;      // 0 or 1
    const int kb   = half * 2;       // K sub-offset within a K=4 slice
    const int lrow = lane >> 1;      // row (0..15) for 4-element LDS ops
    const int lc0  = (lane & 1) * 4; // column base (0 or 4)

    float qerr = 0.f;
    const int gwave = blockIdx.x * 8 + wv;

    for (int it = 0; it < K1_ITERS; ++it) {
        const int tileT = gwave + it * K1_WAVES;   // 0..28671
        const int R0 = tileT * 16;

        // Stage x tile; qsum = 0; residual = x. (float4 per lane: 32*4 = 128)
        {
            const float4 v =
                *reinterpret_cast<const float4*>(x + (size_t)(R0 + lrow) * 8 + lc0);
            *reinterpret_cast<float4*>(&xS[wv][lrow * 8 + lc0]) = v;
            *reinterpret_cast<float4*>(&rS[wv][lrow * 8 + lc0]) = v;
            float4 z; z.x = z.y = z.z = z.w = 0.f;
            *reinterpret_cast<float4*>(&qS[wv][lrow * 8 + lc0]) = z;
        }
        __builtin_amdgcn_wave_barrier();

        for (int lvl = 0; lvl < 2; ++lvl) {
            // A fragments (16x4 f32): lane L -> row L&15, K = 2*(L>>4)+{0,1}
            v2f aLo = *reinterpret_cast<const v2f*>(&rS[wv][col * 8 + kb]);
            v2f aHi = *reinterpret_cast<const v2f*>(&rS[wv][col * 8 + 4 + kb]);

            float bestV[8];
            int   bestI[8];
            #pragma unroll
            for (int r = 0; r < 8; ++r) { bestV[r] = 3.4e38f; bestI[r] = 0; }

            // Sweep 240 roots: 5 fenced chunks x 3 accumulator chains.
            #pragma unroll
            for (int tc = 0; tc < NCHUNKS; ++tc) {
                v2f   bLo[TILE_CHUNK], bHi[TILE_CHUNK];
                float r2v[TILE_CHUNK];
                #pragma unroll
                for (int u = 0; u < TILE_CHUNK; ++u) {
                    const int N = (tc * TILE_CHUNK + u) * 16 + col;
                    bLo[u] = *reinterpret_cast<const v2f*>(&rootsS[N * 8 + kb]);
                    bHi[u] = *reinterpret_cast<const v2f*>(&rootsS[N * 8 + 4 + kb]);
                    r2v[u] = r2S[N];
                }
                v8f acc[TILE_CHUNK];
                #pragma unroll
                for (int u = 0; u < TILE_CHUNK; ++u) {
                    v8f z = {};
                    z = __builtin_amdgcn_wmma_f32_16x16x4_f32(
                            false, aLo, false, bLo[u], (short)0, z, false, false);
                    z = __builtin_amdgcn_wmma_f32_16x16x4_f32(
                            false, aHi, false, bHi[u], (short)0, z, false, false);
                    acc[u] = z;
                }
                #pragma unroll
                for (int u = 0; u < TILE_CHUNK; ++u) {
                    const int N = (tc * TILE_CHUNK + u) * 16 + col;
                    #pragma unroll
                    for (int r = 0; r < 8; ++r) {
                        float sc = r2v[u] - 2.0f * acc[u][r]; // res2 row-const
                        bool take = sc < bestV[r];            // strict: lowest N
                        bestV[r] = take ? sc : bestV[r];
                        bestI[r] = take ? N  : bestI[r];
                    }
                }
                __builtin_amdgcn_sched_barrier(0);  // cap live ranges per chunk
            }

            // Branchless cross-lane argmin: pack (score,index) into one ordered
            // 64-bit key; min64 per step; lowest-index tie-break is implicit.
            unsigned long long k64[8];
            #pragma unroll
            for (int r = 0; r < 8; ++r)
                k64[r] = ((unsigned long long)fkey(bestV[r]) << 32) |
                         (unsigned int)bestI[r];
            #pragma unroll
            for (int m = 1; m < 16; m <<= 1) {
                #pragma unroll
                for (int r = 0; r < 8; ++r) {
                    unsigned long long o = __shfl_xor(k64[r], m, 32);
                    k64[r] = (o < k64[r]) ? o : k64[r];
                }
            }
            if (col == 0) {
                #pragma unroll
                for (int r = 0; r < 8; ++r)
                    idxS[wv][half * 8 + r] = (int)(k64[r] & 0xFFFFFFFFu);
            }
            __builtin_amdgcn_wave_barrier();

            // qsum += roots[idx]; residual = x - qsum  (wave-in-order DS ops)
            {
                const int id = idxS[wv][lrow];
                #pragma unroll
                for (int k = 0; k < 4; ++k) {
                    float q = qS[wv][lrow * 8 + lc0 + k] + rootsS[id * 8 + lc0 + k];
                    qS[wv][lrow * 8 + lc0 + k] = q;
                    rS[wv][lrow * 8 + lc0 + k] = xS[wv][lrow * 8 + lc0 + k] - q;
                }
            }
            __builtin_amdgcn_wave_barrier();

            if (lane < 16) {
                out[IDX_OFF + (size_t)lvl * NROWS + R0 + lane] =
                    (float)idxS[wv][lane];
            }
        }

        // Write quantized (== quantized_ste numerically) + q_err partial.
        {
            float4 qv = *reinterpret_cast<float4*>(&qS[wv][lrow * 8 + lc0]);
            *reinterpret_cast<float4*>(out + (size_t)(R0 + lrow) * 8 + lc0) = qv;
            #pragma unroll
            for (int k = 0; k < 4; ++k) {
                float rr = rS[wv][lrow * 8 + lc0 + k];
                qerr += rr * rr;
            }
        }
    }

    // Deterministic block reduction of q_err partial -> ws[blockIdx].
    redS[tid] = qerr;
    __syncthreads();
    for (int s = 128; s > 0; s >>= 1) {
        if (tid < s) redS[tid] += redS[tid + s];
        __syncthreads();
    }
    if (tid == 0) ws[blockIdx.x] = redS[0];
}

// ---------------------------------------------------------------------------
// Kernel 2: octonion products over Fano lines + alignments + fano partials.
// One thread per (batch, line); 458752 threads exactly.
// ---------------------------------------------------------------------------
__device__ __forceinline__ void qmul(const float* q, const float* p, float* r)
{
    r[0] = q[0]*p[0] - q[1]*p[1] - q[2]*p[2] - q[3]*p[3];
    r[1] = q[0]*p[1] + q[1]*p[0] + q[2]*p[3] - q[3]*p[2];
    r[2] = q[0]*p[2] - q[1]*p[3] + q[2]*p[0] + q[3]*p[1];
    r[3] = q[0]*p[3] + q[1]*p[2] - q[2]*p[1] + q[3]*p[0];
}

__global__ __launch_bounds__(256)
void fano_kernel(const float* __restrict__ quant,
                 const int* __restrict__ li, const int* __restrict__ lj,
                 const int* __restrict__ lk,
                 float* __restrict__ out, float* __restrict__ wsf)
{
    __shared__ float redS[256];
    const int g  = blockIdx.x * 256 + threadIdx.x;   // 0..458751
    const int b  = g / 7;
    const int ln = g - b * 7;
    const int i = li[ln], j = lj[ln], k = lk[ln];

    float X[8], Y[8], C[8];
    #pragma unroll
    for (int e = 0; e < 8; ++e) {
        X[e] = quant[(size_t)b * 56 + i * 8 + e];
        Y[e] = quant[(size_t)b * 56 + j * 8 + e];
        C[e] = quant[(size_t)b * 56 + k * 8 + e];
    }

    // oct_mul(X, Y): X=(a,bq), Y=(c,dq)
    const float* a  = X;
    const float* bq = X + 4;
    const float* c  = Y;
    const float* dq = Y + 4;
    float cConj[4] = { c[0], -c[1], -c[2], -c[3] };
    float dConj[4] = { dq[0], -dq[1], -dq[2], -dq[3] };

    float t1[4], t2[4], t3[4], t4[4], P[8];
    qmul(a, c, t1);
    qmul(dConj, bq, t2);
    qmul(dq, a, t3);
    qmul(bq, cConj, t4);
    #pragma unroll
    for (int e = 0; e < 4; ++e) {
        P[e]     = t1[e] - t2[e];
        P[e + 4] = t3[e] + t4[e];
    }

    float dot = 0.f, np = 0.f, nc = 0.f;
    #pragma unroll
    for (int e = 0; e < 8; ++e) {
        dot += P[e] * C[e];
        np  += P[e] * P[e];
        nc  += C[e] * C[e];
        out[PROD_OFF + (size_t)g * 8 + e] = P[e];
    }
    float denom = fmaxf(sqrtf(np) * sqrtf(nc), 1e-8f);
    float align = dot / denom;
    out[ALIGN_OFF + (size_t)g] = align;

    float contrib = 1.0f - fminf(fmaxf(align, -1.0f), 1.0f);

    redS[threadIdx.x] = contrib;
    __syncthreads();
    for (int s = 128; s > 0; s >>= 1) {
        if (threadIdx.x < s) redS[threadIdx.x] += redS[threadIdx.x + s];
        __syncthreads();
    }
    if (threadIdx.x == 0) wsf[blockIdx.x] = redS[0];
}

// ---------------------------------------------------------------------------
// Kernel 3: deterministic serial finalize of both scalars.
// ---------------------------------------------------------------------------
__global__ void finalize_kernel(const float* __restrict__ ws,
                                float* __restrict__ out)
{
    if (threadIdx.x == 0) {
        float s = 0.f;
        for (int i = 0; i < K1_BLOCKS; ++i) s += ws[i];
        out[QERR_OFF] = s / 3670016.0f;          // mean over B*7*8
    } else if (threadIdx.x == 1) {
        float s = 0.f;
        for (int i = 0; i < K2_BLOCKS; ++i) s += ws[K1_BLOCKS + i];
        float f = 0.5f * (s / 458752.0f);        // mean over B*7, * 0.5
        out[FANO_OFF] = fminf(fmaxf(f, 0.0f), 1.0f);
    }
}

extern "C" void kernel_launch(void* const* d_in, const int* in_sizes, int n_in,
                              void* d_out, int out_size, void* d_ws, size_t ws_size,
                              hipStream_t stream)
{
    (void)in_sizes; (void)n_in; (void)out_size; (void)ws_size;
    const float* states = (const float*)d_in[0];
    const float* roots  = (const float*)d_in[1];
    const int*   li     = (const int*)d_in[2];
    const int*   lj     = (const int*)d_in[3];
    const int*   lk     = (const int*)d_in[4];
    float* out = (float*)d_out;
    float* ws  = (float*)d_ws;

    hipLaunchKernelGGL(quant_kernel, dim3(K1_BLOCKS), dim3(256), 0, stream,
                       states, roots, out, ws);
    hipLaunchKernelGGL(fano_kernel, dim3(K2_BLOCKS), dim3(256), 0, stream,
                       out, li, lj, lk, out, ws + K1_BLOCKS);
    hipLaunchKernelGGL(finalize_kernel, dim3(1), dim3(64), 0, stream, ws, out);
}